// PackedViterbi_22514218566008
// MI455X (gfx1250) — compile-verified
//
#include <hip/hip_runtime.h>
#include <stdint.h>

#define S_DIM 64
#define RING_D 8          // prefetch depth (power of two); 8 * 16KB = 128KB LDS ring
#define TPB 256           // 4 threads per row * 64 rows

typedef int v4i __attribute__((ext_vector_type(4)));
typedef __attribute__((address_space(1))) v4i* g4p_t;   // global b128 pointer
typedef __attribute__((address_space(3))) v4i* l4p_t;   // LDS b128 pointer

// ---------------- CDNA5 async global->LDS copy (ASYNCcnt path) ----------------
#if __has_builtin(__builtin_amdgcn_global_load_async_to_lds_b128)
#define HAVE_ASYNC_BUILTIN 1
#endif

__device__ __forceinline__ void async_copy_64B(const float* g, float* lds) {
#if defined(HAVE_ASYNC_BUILTIN)
  // (global v4i* AS1, lds v4i* AS3, imm offset applied to both, cpol)
  g4p_t gp = (g4p_t)(g);
  l4p_t lp = (l4p_t)(lds);
  __builtin_amdgcn_global_load_async_to_lds_b128(gp, lp, 0, 0);
  __builtin_amdgcn_global_load_async_to_lds_b128(gp, lp, 16, 0);
  __builtin_amdgcn_global_load_async_to_lds_b128(gp, lp, 32, 0);
  __builtin_amdgcn_global_load_async_to_lds_b128(gp, lp, 48, 0);
#else
  uint32_t lo = (uint32_t)(uintptr_t)(__attribute__((address_space(3))) void*)(lds);
  uint64_t ga = (uint64_t)(uintptr_t)g;
  asm volatile(
      "global_load_async_to_lds_b128 %0, %1, off\n\t"
      "global_load_async_to_lds_b128 %0, %1, off offset:16\n\t"
      "global_load_async_to_lds_b128 %0, %1, off offset:32\n\t"
      "global_load_async_to_lds_b128 %0, %1, off offset:48"
      :: "v"(lo), "v"(ga) : "memory");
#endif
}

#if __has_builtin(__builtin_amdgcn_s_wait_asynccnt)
#define WAIT_ASYNC(n) __builtin_amdgcn_s_wait_asynccnt(n)
#else
#define WAIT_ASYNC(n) asm volatile("s_wait_asynccnt %0" :: "i"(n) : "memory")
#endif

// ---------------- fast trans ops (v_exp_f32 / v_log_f32 are base-2) ----------
__device__ __forceinline__ float fexp2(float x) {
#if __has_builtin(__builtin_amdgcn_exp2f)
  return __builtin_amdgcn_exp2f(x);
#else
  return exp2f(x);
#endif
}
__device__ __forceinline__ float flog2(float x) {
#if __has_builtin(__builtin_amdgcn_logf)
  return __builtin_amdgcn_logf(x);
#else
  return log2f(x);
#endif
}

#define LOG2E 1.4426950408889634f
#define LN2   0.6931471805599453f

// One workgroup per batch element b. Thread tid -> row i = tid>>2, j-quarter
// q = tid&3 (16 contiguous j). Each thread's 64B global slice (tid*64 bytes of
// the 16KB step tile) is exactly the slice it consumes, so theta staging needs
// only the wave's own ASYNCcnt wait; the single per-step barrier protects the
// double-buffered state vector.
//
// State kept in base-2 log domain: W = V * log2(e). Inner op is then a single
// FMA per element: y = theta*LOG2E + W[j]; W_new[i] = max_j y + log2(sum 2^(y-max)).
__global__ __launch_bounds__(TPB, 1) void packed_lse_viterbi(
    const float* __restrict__ theta,   // [T, B, S, S]
    const int*   __restrict__ lengths, // [B]
    float*       __restrict__ out,     // [B]
    int T, int B) {
  extern __shared__ float smem[];      // [RING_D * S*S] theta ring + [2*S] W
  const int b   = (int)blockIdx.x;
  const int tid = (int)threadIdx.x;
  const int i   = tid >> 2;
  const int q   = tid & 3;
  const int j0  = q << 4;

  float* Wr = smem + RING_D * (S_DIM * S_DIM);  // W read buffer
  float* Ww = Wr + S_DIM;                       // W write buffer

  if (tid < S_DIM) Wr[tid] = 0.0f;              // V0 = 0 -> W0 = 0

  const int    len        = lengths[b];
  const size_t stepStride = (size_t)B * S_DIM * S_DIM;
  const float* gbase      = theta + (size_t)b * S_DIM * S_DIM + (size_t)tid * 16;

  // Prime the ring: steps 0..RING_D-1 in flight (4 b128 async loads each).
  #pragma unroll
  for (int p = 0; p < RING_D; ++p)
    async_copy_64B(gbase + (size_t)p * stepStride,
                   smem + p * (S_DIM * S_DIM) + tid * 16);

  __syncthreads();  // W0 visible

  for (int t = 0; t < len; ++t) {
    // This wave's oldest 4 async loads (step t's tile slice) are complete.
    WAIT_ASYNC(4 * (RING_D - 1));

    const int buf = t & (RING_D - 1);
    const float4* thv = (const float4*)(smem + buf * (S_DIM * S_DIM) + tid * 16);
    const float4* wv  = (const float4*)(Wr + j0);

    float y[16];
    float m = -3.402823466e38f;
    #pragma unroll
    for (int k = 0; k < 4; ++k) {
      float4 tq = thv[k];              // ds_load_b128
      float4 wq = wv[k];               // ds_load_b128
      y[4*k+0] = fmaf(tq.x, LOG2E, wq.x);   // v_fma_f32
      y[4*k+1] = fmaf(tq.y, LOG2E, wq.y);
      y[4*k+2] = fmaf(tq.z, LOG2E, wq.z);
      y[4*k+3] = fmaf(tq.w, LOG2E, wq.w);
      m = fmaxf(fmaxf(m, fmaxf(y[4*k+0], y[4*k+1])), fmaxf(y[4*k+2], y[4*k+3]));
    }

    // theta consumed into regs: refill the freed ring slot with step t+RING_D
    // now so the async issue overlaps the remaining math (clamped in-bounds,
    // unused past len; exactly 4 issues/iter keeps ASYNCcnt bookkeeping fixed).
    int tf = t + RING_D; if (tf > T - 1) tf = T - 1;
    async_copy_64B(gbase + (size_t)tf * stepStride,
                   smem + buf * (S_DIM * S_DIM) + tid * 16);

    // row max across the 4 lanes that share row i
    m = fmaxf(m, __shfl_xor(m, 1, 32));
    m = fmaxf(m, __shfl_xor(m, 2, 32));

    float s = 0.0f;
    #pragma unroll
    for (int k = 0; k < 16; ++k) s += fexp2(y[k] - m);
    s += __shfl_xor(s, 1, 32);
    s += __shfl_xor(s, 2, 32);

    if (q == 0) Ww[i] = m + flog2(s);

    __syncthreads();                   // W_new visible to all waves
    float* tmp = Wr; Wr = Ww; Ww = tmp;
  }

  // Terminal: out[b] = ln2 * log2(sum_i 2^(W_final[i])), done by wave 0.
  if (tid < 32) {
    float a = Wr[tid];
    float c = Wr[tid + 32];
    float m = fmaxf(a, c);
    float s = fexp2(a - m) + fexp2(c - m);
    #pragma unroll
    for (int off = 16; off > 0; off >>= 1) {
      float mo = __shfl_xor(m, off, 32);
      float so = __shfl_xor(s, off, 32);
      float mn = fmaxf(m, mo);
      s = s * fexp2(m - mn) + so * fexp2(mo - mn);
      m = mn;
    }
    if (tid == 0) out[b] = LN2 * (m + flog2(s));
  }
}

extern "C" void kernel_launch(void* const* d_in, const int* in_sizes, int n_in,
                              void* d_out, int out_size, void* d_ws, size_t ws_size,
                              hipStream_t stream) {
  (void)n_in; (void)d_ws; (void)ws_size; (void)out_size;
  const float* theta   = (const float*)d_in[0];
  const int*   lengths = (const int*)d_in[1];
  float*       out     = (float*)d_out;

  const int B = in_sizes[1];                         // 64
  const int T = in_sizes[0] / (B * S_DIM * S_DIM);   // 512

  const size_t shmem = (size_t)(RING_D * S_DIM * S_DIM + 2 * S_DIM) * sizeof(float);
  // Opt in to >64KB dynamic LDS (320KB/WGP on CDNA5).
  (void)hipFuncSetAttribute((const void*)packed_lse_viterbi,
                            hipFuncAttributeMaxDynamicSharedMemorySize, (int)shmem);

  packed_lse_viterbi<<<B, TPB, shmem, stream>>>(theta, lengths, out, T, B);
}